// VQLayer_72095321031031
// MI455X (gfx1250) — compile-verified
//
#include <hip/hip_runtime.h>
#include <hip/hip_bf16.h>

typedef float v2f __attribute__((ext_vector_type(2)));
typedef float v4f __attribute__((ext_vector_type(4)));
typedef float v8f __attribute__((ext_vector_type(8)));

#define VQ_D      128
#define VQ_K      1024
#define VQ_N      32768        // 32 * 1024 rows
#define KSTEPS    32           // 128 / 4 (K=4 per f32 WMMA)
#define CTILES    64           // 1024 / 16 code tiles
#define NTILES    2048         // 32768 / 16 row tiles

// ---------------------------------------------------------------------------
// Kernel A: codebook squared norms. One wave (32 lanes) per code row.
// ---------------------------------------------------------------------------
__global__ void __launch_bounds__(256)
vq_cnorm_kernel(const float* __restrict__ cb, float* __restrict__ cnorm) {
    const int wave = (blockIdx.x * blockDim.x + threadIdx.x) >> 5;  // code id
    const int lane = threadIdx.x & 31;
    const float* row = cb + wave * VQ_D;
    v4f v = *(const v4f*)(row + lane * 4);
    float s = v.x * v.x + v.y * v.y + v.z * v.z + v.w * v.w;
    #pragma unroll
    for (int m = 16; m >= 1; m >>= 1) s += __shfl_xor(s, m, 32);
    if (lane == 0) cnorm[wave] = s;
}

// ---------------------------------------------------------------------------
// Kernel B: WMMA distance argmin + gather + z_q + per-tile SSE partials.
// One wave handles one 16-row tile of z against all 1024 codes.
// ---------------------------------------------------------------------------
__global__ void __launch_bounds__(256)
vq_main_kernel(const float* __restrict__ z, const float* __restrict__ cb,
               const float* __restrict__ cnorm,
               float* __restrict__ zq, float* __restrict__ idx_out,
               float* __restrict__ sse_out) {
    const int lane    = threadIdx.x & 31;
    const int wib     = threadIdx.x >> 5;            // wave in block (0..7)
    const int tile    = blockIdx.x * 8 + wib;        // row tile (0..2047)
    const int rowBase = tile * 16;
    const int half    = lane >> 4;                   // 0: lanes 0-15, 1: 16-31
    const int l16     = lane & 15;

    // ---- Load A tile (16x128 f32) into registers, ISA 16x4 f32 A layout:
    // lane = row M, per-WMMA K pair = {2*half, 2*half+1}  -> float2 loads.
    const float* zrow = z + (size_t)(rowBase + l16) * VQ_D + 2 * half;
    v2f a[KSTEPS];
    #pragma unroll
    for (int kk = 0; kk < KSTEPS; ++kk)
        a[kk] = *(const v2f*)(zrow + 4 * kk);

    // Running (min, argmin) per C-matrix register j (row = j + 8*half).
    float best[8];
    int   bidx[8];
    #pragma unroll
    for (int j = 0; j < 8; ++j) { best[j] = 3.4e38f; bidx[j] = 0; }

    // ---- Sweep code tiles, two at a time for ILP across the matrix pipe.
    for (int ct = 0; ct < CTILES; ct += 2) {
        const int n0 = (ct + 0) * 16 + l16;          // code id this lane scores
        const int n1 = (ct + 1) * 16 + l16;
        // B layout (4x16 f32): lane = column n, K pair = {2*half, 2*half+1}.
        const float* b0p = cb + (size_t)n0 * VQ_D + 2 * half;
        const float* b1p = cb + (size_t)n1 * VQ_D + 2 * half;
        const float cn0 = cnorm[n0];
        const float cn1 = cnorm[n1];

        v8f c0 = {0.f, 0.f, 0.f, 0.f, 0.f, 0.f, 0.f, 0.f};
        v8f c1 = {0.f, 0.f, 0.f, 0.f, 0.f, 0.f, 0.f, 0.f};
        #pragma unroll
        for (int kk = 0; kk < KSTEPS; ++kk) {
            v2f b0 = *(const v2f*)(b0p + 4 * kk);
            v2f b1 = *(const v2f*)(b1p + 4 * kk);
            c0 = __builtin_amdgcn_wmma_f32_16x16x4_f32(
                     false, a[kk], false, b0, (short)0, c0, false, false);
            c1 = __builtin_amdgcn_wmma_f32_16x16x4_f32(
                     false, a[kk], false, b1, (short)0, c1, false, false);
        }

        // dist' = ||c||^2 - 2*(z.c)   (||z||^2 dropped: constant per row).
        // Strict < keeps the earliest index on ties (n0 < n1, ct ascending).
        #pragma unroll
        for (int j = 0; j < 8; ++j) {
            const float d0 = cn0 - 2.0f * c0[j];
            const bool  t0 = d0 < best[j];
            best[j] = t0 ? d0 : best[j];
            bidx[j] = t0 ? n0 : bidx[j];
            const float d1 = cn1 - 2.0f * c1[j];
            const bool  t1 = d1 < best[j];
            best[j] = t1 ? d1 : best[j];
            bidx[j] = t1 ? n1 : bidx[j];
        }
    }

    // ---- Cross-lane (min, idx) reduction within each 16-lane half.
    // Rows 0..7 live in lanes 0-15, rows 8..15 in lanes 16-31 (C layout).
    #pragma unroll
    for (int j = 0; j < 8; ++j) {
        float v  = best[j];
        int   bi = bidx[j];
        #pragma unroll
        for (int m = 8; m >= 1; m >>= 1) {
            const float ov = __shfl_xor(v,  m, 32);
            const int   oi = __shfl_xor(bi, m, 32);
            const bool take = (ov < v) || (ov == v && oi < bi);
            v  = take ? ov : v;
            bi = take ? oi : bi;
        }
        best[j] = v;
        bidx[j] = bi;
    }

    // ---- Emit indices (as float, per d_out convention).
    if (l16 == 0) {
        #pragma unroll
        for (int j = 0; j < 8; ++j)
            idx_out[rowBase + half * 8 + j] = (float)bidx[j];
    }

    // ---- Gather winners, write z_q (forward value == quantized), SSE.
    float sse = 0.0f;
    #pragma unroll
    for (int m = 0; m < 16; ++m) {
        const int qi = __shfl(bidx[m & 7], (m >> 3) * 16, 32);
        const v4f q  = *(const v4f*)(cb + (size_t)qi * VQ_D + lane * 4);
        const v4f zv = *(const v4f*)(z + (size_t)(rowBase + m) * VQ_D + lane * 4);
        const v4f dq = q - zv;
        sse += dq.x * dq.x + dq.y * dq.y + dq.z * dq.z + dq.w * dq.w;
        *(v4f*)(zq + (size_t)(rowBase + m) * VQ_D + lane * 4) = q;
    }
    #pragma unroll
    for (int m = 16; m >= 1; m >>= 1) sse += __shfl_xor(sse, m, 32);
    if (lane == 0) sse_out[tile] = sse;
}

// ---------------------------------------------------------------------------
// Kernel C: deterministic fixed-order reduction of 2048 partials -> vq_loss.
// vq_loss = (1 + 0.25) * mean((q - z)^2)   (both loss terms share the value).
// ---------------------------------------------------------------------------
__global__ void __launch_bounds__(256)
vq_loss_kernel(const float* __restrict__ sse, float* __restrict__ loss) {
    __shared__ float red[256];
    float s = 0.0f;
    for (int i = threadIdx.x; i < NTILES; i += 256) s += sse[i];
    red[threadIdx.x] = s;
    __syncthreads();
    for (int off = 128; off >= 1; off >>= 1) {
        if (threadIdx.x < off) red[threadIdx.x] += red[threadIdx.x + off];
        __syncthreads();
    }
    if (threadIdx.x == 0)
        loss[0] = 1.25f * red[0] / (float)(VQ_N * VQ_D);
}

// ---------------------------------------------------------------------------
extern "C" void kernel_launch(void* const* d_in, const int* in_sizes, int n_in,
                              void* d_out, int out_size, void* d_ws, size_t ws_size,
                              hipStream_t stream) {
    const float* z  = (const float*)d_in[0];   // [32,1024,128] f32
    const float* cb = (const float*)d_in[1];   // [1024,128]   f32

    float* out  = (float*)d_out;
    float* zq   = out;                          // 4194304 floats
    float* loss = out + (size_t)VQ_N * VQ_D;    // 1 float
    float* idx  = loss + 1;                     // 32768 floats (indices)

    float* cnorm = (float*)d_ws;                // 1024 floats
    float* sse   = cnorm + VQ_K;                // 2048 floats

    vq_cnorm_kernel<<<VQ_K / 8, 256, 0, stream>>>(cb, cnorm);
    vq_main_kernel<<<NTILES / 8, 256, 0, stream>>>(z, cb, cnorm, zq, idx, sse);
    vq_loss_kernel<<<1, 256, 0, stream>>>(sse, loss);
}